// HighOrderAggregator_89404039233610
// MI455X (gfx1250) — compile-verified
//
#include <hip/hip_runtime.h>
#include <hip/hip_bf16.h>

typedef __attribute__((ext_vector_type(2))) float v2f;
typedef __attribute__((ext_vector_type(8))) float v8f;

#define D 128           // D_IN == D_OUT
#define NT 8            // 128 / 16 column tiles
#define KSTEPS 32       // 128 / 4 k-steps for f32 WMMA
#define WAVES_PER_WG 8  // 256 threads -> 8x reuse of the LDS-staged W

// ---------------------------------------------------------------------------
// Zero workspace (hop1 + hop2) — d_ws is poisoned, must clear every call.
// ---------------------------------------------------------------------------
__global__ void hoa_zero_kernel(float4* __restrict__ p, long n4) {
    long i = (long)blockIdx.x * blockDim.x + threadIdx.x;
    long stride = (long)gridDim.x * blockDim.x;
    for (; i < n4; i += stride) p[i] = make_float4(0.f, 0.f, 0.f, 0.f);
}

// ---------------------------------------------------------------------------
// Scatter SpMM: y[row[e]] += vals[e] * x[col[e]]   (one wave per edge)
// Lanes cover the 128-wide feature row as 32 x float4; 4 f32 atomics per lane.
// x / y are L2-resident (51.2 MB each vs 192 MB L2).
// ---------------------------------------------------------------------------
__global__ __launch_bounds__(256) void hoa_spmm_scatter(
    const float* __restrict__ x, const float* __restrict__ vals,
    const int* __restrict__ row, const int* __restrict__ col,
    float* __restrict__ y, int E) {
    int gid  = blockIdx.x * blockDim.x + threadIdx.x;
    int e    = gid >> 5;
    int lane = gid & 31;
    if (e >= E) return;
    int   r = row[e];
    int   c = col[e];
    float v = vals[e];
    const float4 xv = *(const float4*)(x + (size_t)c * D + lane * 4);
    float* dst = y + (size_t)r * D + lane * 4;
    atomicAdd(dst + 0, xv.x * v);
    atomicAdd(dst + 1, xv.y * v);
    atomicAdd(dst + 2, xv.z * v);
    atomicAdd(dst + 3, xv.w * v);
}

// ---------------------------------------------------------------------------
// LDS layout for W (swizzled for WMMA B operand, 8192 x float2 = 64 KB):
//   sWp[(kk*8 + nt)*32 + half*16 + lm] = { W[4kk+2half][nt*16+lm],
//                                          W[4kk+2half+1][nt*16+lm] }
// One B fetch = one ds_load_b64 into an even-aligned VGPR pair (WMMA-ready);
// the 32 lanes of a wave cover one contiguous 256B block = all 64 banks.
// (Compiler fuses adjacent nt fetches into ds_load_2addr_b64.)
// ---------------------------------------------------------------------------
__device__ __forceinline__ void hoa_stage_w_swz(v2f* sWp, const float* __restrict__ W,
                                                int tid, int nthreads) {
    // 2048 groups; each group = 4 consecutive lm values (8 W elements).
#pragma unroll
    for (int g = tid; g < 2048; g += nthreads) {
        int lmq  = g & 3;          // lm = lmq*4 .. lmq*4+3
        int half = (g >> 2) & 1;
        int nt   = (g >> 3) & 7;
        int kk   = g >> 6;
        int k  = 4 * kk + 2 * half;
        int n0 = nt * 16 + lmq * 4;
        float4 a = *(const float4*)(W + (size_t)k * D + n0);        // row k
        float4 b = *(const float4*)(W + (size_t)(k + 1) * D + n0);  // row k+1
        float4* d4 = (float4*)(sWp + (size_t)kk * 256 + nt * 32 + half * 16 + lmq * 4);
        d4[0] = make_float4(a.x, b.x, a.y, b.y);
        d4[1] = make_float4(a.z, b.z, a.w, b.w);
    }
}

// ---------------------------------------------------------------------------
// One hop of the fused GEMM: t = relu(X @ W + b);
// outacc += sc * (t - mean_row) * rsqrt(var_row + eps) + off
// ---------------------------------------------------------------------------
__device__ __forceinline__ void hoa_gemm_hop(
    const float* __restrict__ X, const v2f* sWp,
    const float* __restrict__ bias, const float* __restrict__ off,
    const float* __restrict__ sc,
    int m0, int half, int lm, v8f outacc[NT]) {

    v8f acc[NT] = {};

    // A operand: lane l -> M = lm, K = 4*kk + 2*half + {0,1}  (contiguous float2)
    const float* arow = X + (size_t)(m0 + lm) * D + 2 * half;

    for (int kk = 0; kk < KSTEPS; ++kk) {
        v2f a = *(const v2f*)(arow + 4 * kk);
        const v2f* wbase = sWp + (size_t)kk * 256 + half * 16 + lm;
#pragma unroll
        for (int nt = 0; nt < NT; ++nt) {
            v2f b = wbase[nt * 32];   // ds_load_b64, even-aligned pair
            acc[nt] = __builtin_amdgcn_wmma_f32_16x16x4_f32(
                false, a, false, b, (short)0, acc[nt], false, false);
        }
    }

    // bias + relu (lane's column n = nt*16 + lm is the same for all 8 rows)
#pragma unroll
    for (int nt = 0; nt < NT; ++nt) {
        float bv = bias[nt * 16 + lm];
#pragma unroll
        for (int r = 0; r < 8; ++r) {
            float t = acc[nt][r] + bv;
            acc[nt][r] = t > 0.f ? t : 0.f;
        }
    }

    // per-row mean / var over 128 columns.
    // VGPR r holds row (r + 8*half); reduce across the 16 lanes of each half
    // with xor masks 1,2,4,8 (stay inside the half-group).
    float s[8], s2[8];
#pragma unroll
    for (int r = 0; r < 8; ++r) {
        float a0 = 0.f, a1 = 0.f;
#pragma unroll
        for (int nt = 0; nt < NT; ++nt) {
            float v = acc[nt][r];
            a0 += v;
            a1 += v * v;
        }
        s[r] = a0; s2[r] = a1;
    }
#pragma unroll
    for (int m = 1; m <= 8; m <<= 1) {
#pragma unroll
        for (int r = 0; r < 8; ++r) {
            s[r]  += __shfl_xor(s[r],  m, 32);
            s2[r] += __shfl_xor(s2[r], m, 32);
        }
    }
    float mean[8], rstd[8];
#pragma unroll
    for (int r = 0; r < 8; ++r) {
        float mu = s[r] * (1.0f / D);
        float var = s2[r] * (1.0f / D) - mu * mu;
        mean[r] = mu;
        rstd[r] = rsqrtf(var + 1e-9f);
    }

    // scale/offset + accumulate into the running output sum
#pragma unroll
    for (int nt = 0; nt < NT; ++nt) {
        float scv  = sc[nt * 16 + lm];
        float offv = off[nt * 16 + lm];
#pragma unroll
        for (int r = 0; r < 8; ++r) {
            outacc[nt][r] += scv * (acc[nt][r] - mean[r]) * rstd[r] + offv;
        }
    }
}

__global__ __launch_bounds__(256) void hoa_gemm_fused(
    const float* __restrict__ hop0, const float* __restrict__ hop1,
    const float* __restrict__ hop2,
    const float* __restrict__ W0, const float* __restrict__ b0,
    const float* __restrict__ off0, const float* __restrict__ sc0,
    const float* __restrict__ W1, const float* __restrict__ b1,
    const float* __restrict__ off1, const float* __restrict__ sc1,
    const float* __restrict__ W2, const float* __restrict__ b2,
    const float* __restrict__ off2, const float* __restrict__ sc2,
    float* __restrict__ out, int nblocks) {

    __shared__ v2f sWp[D * D / 2];   // 64 KB swizzled W (WGP has 320 KB LDS)

    int tid  = threadIdx.x;
    int wave = tid >> 5;
    int lane = tid & 31;
    int rb = blockIdx.x * WAVES_PER_WG + wave;
    // nblocks (6250) is not a multiple of 8: inactive waves must still hit the
    // barriers, so guard compute (wave-uniform => EXEC all-1s inside WMMA).
    bool active = rb < nblocks;

    int m0   = rb * 16;
    int half = lane >> 4;   // 0: rows 0-7, 1: rows 8-15 of the C/D layout
    int lm   = lane & 15;   // column within a 16-wide tile

    v8f outacc[NT] = {};

    // Explicitly unrolled hop sequence: direct __restrict__ pointers keep the
    // A-operand loads on the pure GLOBAL path (LOADcnt only) instead of FLAT
    // (which also ties up DScnt / the LDS issue path we need for B operands).
    hoa_stage_w_swz(sWp, W0, tid, blockDim.x);
    __syncthreads();
    if (active) hoa_gemm_hop(hop0, sWp, b0, off0, sc0, m0, half, lm, outacc);
    __syncthreads();

    hoa_stage_w_swz(sWp, W1, tid, blockDim.x);
    __syncthreads();
    if (active) hoa_gemm_hop(hop1, sWp, b1, off1, sc1, m0, half, lm, outacc);
    __syncthreads();

    hoa_stage_w_swz(sWp, W2, tid, blockDim.x);
    __syncthreads();
    if (active) hoa_gemm_hop(hop2, sWp, b2, off2, sc2, m0, half, lm, outacc);

    if (active) {
        // store: element (row = m0 + r + 8*half, col = nt*16 + lm)
#pragma unroll
        for (int nt = 0; nt < NT; ++nt) {
#pragma unroll
            for (int r = 0; r < 8; ++r) {
                out[(size_t)(m0 + r + 8 * half) * D + nt * 16 + lm] = outacc[nt][r];
            }
        }
    }
}

// ---------------------------------------------------------------------------
extern "C" void kernel_launch(void* const* d_in, const int* in_sizes, int n_in,
                              void* d_out, int out_size, void* d_ws, size_t ws_size,
                              hipStream_t stream) {
    const float* vecs = (const float*)d_in[0];
    const float* vals = (const float*)d_in[1];
    const int*   row  = (const int*)d_in[2];
    const int*   col  = (const int*)d_in[3];
    const float* W0 = (const float*)d_in[4];
    const float* b0 = (const float*)d_in[5];
    const float* off0 = (const float*)d_in[6];
    const float* sc0  = (const float*)d_in[7];
    const float* W1 = (const float*)d_in[8];
    const float* b1 = (const float*)d_in[9];
    const float* off1 = (const float*)d_in[10];
    const float* sc1  = (const float*)d_in[11];
    const float* W2 = (const float*)d_in[12];
    const float* b2 = (const float*)d_in[13];
    const float* off2 = (const float*)d_in[14];
    const float* sc2  = (const float*)d_in[15];
    float* out = (float*)d_out;

    const int N = in_sizes[0] / D;     // 100000
    const int E = in_sizes[1];         // 1600000

    float* hop1 = (float*)d_ws;
    float* hop2 = hop1 + (size_t)N * D;

    // 1) zero hop buffers (2 * N * 128 floats)
    long n4 = (long)2 * N * D / 4;
    int zblocks = (int)((n4 + 255) / 256);
    if (zblocks > 16384) zblocks = 16384;
    hoa_zero_kernel<<<zblocks, 256, 0, stream>>>((float4*)d_ws, n4);

    // 2) hop1 = A @ vecs
    long spmm_threads = (long)E * 32;
    int sblocks = (int)((spmm_threads + 255) / 256);
    hoa_spmm_scatter<<<sblocks, 256, 0, stream>>>(vecs, vals, row, col, hop1, E);

    // 3) hop2 = A @ hop1
    hoa_spmm_scatter<<<sblocks, 256, 0, stream>>>(hop1, vals, row, col, hop2, E);

    // 4) fused triple GEMM + relu + rownorm + sum (swizzled W staged in LDS)
    int nblocks = N / 16;                       // 6250 (N divisible by 16)
    int gblocks = (nblocks + WAVES_PER_WG - 1) / WAVES_PER_WG;   // 782
    hoa_gemm_fused<<<gblocks, 256, 0, stream>>>(
        vecs, hop1, hop2,
        W0, b0, off0, sc0,
        W1, b1, off1, sc1,
        W2, b2, off2, sc2,
        out, nblocks);
}